// DeepHOG_4346506903849
// MI455X (gfx1250) — compile-verified
//
#include <hip/hip_runtime.h>
#include <math.h>

typedef __attribute__((ext_vector_type(2))) float v2f;
typedef __attribute__((ext_vector_type(8))) float v8f;

#define NBATCH 64
#define HH 1024
#define WW 1024
#define NBX 64          // 16x16 blocks per image row
#define NBLK 4096       // blocks per image

// flat f32 offsets into d_out (reference return order)
#define OUT_OUTPUT 0ull
#define OUT_BLOCKS 1048576ull                  // 64*4096*4
#define OUT_MU     68157440ull                 // + 64*4096*256
#define OUT_U      68157568ull                 // + 64*2
#define OUT_THETA  68157696ull                 // + 64*2

// f32 offsets into workspace
#define WS_ACC 0            // 64 batches * 8 (S, Sx, Sy, Sxx, Syy, Sxy, pad, pad)
#define WS_REC (64*8)       // 64*4096 records * 8 (mux,muy,vx,vy,L1,L2,bmean,pad)

__global__ void k_zero_acc(float* __restrict__ ws) {
    int t = blockIdx.x * blockDim.x + threadIdx.x;
    if (t < 64 * 8) ws[WS_ACC + t] = 0.0f;
}

__device__ __forceinline__ void wait_async0() {
#if __has_builtin(__builtin_amdgcn_s_wait_asynccnt)
    __builtin_amdgcn_s_wait_asynccnt(0);
#else
    asm volatile("s_wait_asynccnt 0x0" ::: "memory");
#endif
}

__global__ __launch_bounds__(256) void k_blocks(const float* __restrict__ x,
                                                float* __restrict__ out,
                                                float* __restrict__ ws) {
    __shared__ float tile[16][132];   // 16 rows x 128 cols (+4 pad)

    const int wg   = blockIdx.x;      // 64 batches * 64 block-rows * 8 groups
    const int b    = wg >> 9;
    const int rem  = wg & 511;
    const int brow = rem >> 3;
    const int bgrp = rem & 7;
    const int tid  = threadIdx.x;

    // ---- async DMA of 16x128 tile into LDS (ASYNCcnt path) ----
    const float* src = x + (size_t)b * (HH * WW) + (size_t)(brow * 16) * WW + bgrp * 128;
#pragma unroll
    for (int j = 0; j < 2; ++j) {
        int f4 = j * 256 + tid;
        int r  = f4 >> 5;
        int c4 = (f4 & 31) << 2;
        const float* g = src + (size_t)r * WW + c4;
        unsigned lds = (unsigned)(size_t)(&tile[r][c4]);   // low 32 bits = LDS byte addr
        asm volatile("global_load_async_to_lds_b128 %0, %1, off"
                     :: "v"(lds), "v"(g) : "memory");
    }
    wait_async0();
    __syncthreads();

    const int wid  = tid >> 5;            // wave 0..7 -> one 16x16 block each
    const int lane = tid & 31;
    const int l    = brow * NBX + bgrp * 8 + wid;
    const int cb   = wid * 16;

    // ---- blocks output: 1KB contiguous per wave ----
    {
        float* dst = out + OUT_BLOCKS + ((size_t)b * NBLK + l) * 256 + lane * 8;
        int r = lane >> 1, c = (lane & 1) * 8;
        float4 v0 = *reinterpret_cast<float4*>(&tile[r][cb + c]);
        float4 v1 = *reinterpret_cast<float4*>(&tile[r][cb + c + 4]);
        *reinterpret_cast<float4*>(dst)     = v0;
        *reinterpret_cast<float4*>(dst + 4) = v1;
    }

    const int m = lane & 15;              // A: row / B: column index
    const int h = lane >> 4;              // lane half -> K slots {2h, 2h+1}

    // Branchless basis weights: basis(m, t) = w0 + t*(w1 + w2*t)
    const float w0 = (m == 0) ? 1.0f : 0.0f;
    const float w1 = (m == 1) ? 1.0f : 0.0f;
    const float w2 = (m == 2) ? 1.0f : 0.0f;
    const float th2 = 2.0f * (float)h;

    // Sample points are identical for stage1 (x basis) and stage2 (y basis):
    // t0 = 4i + 2h, t1 = t0 + 1. Precompute the 4 pairs once.
    v2f bas[4];
#pragma unroll
    for (int i = 0; i < 4; ++i) {
        float t0 = (float)(4 * i) + th2;
        float t1 = t0 + 1.0f;
        bas[i].x = w0 + t0 * (w1 + w2 * t0);
        bas[i].y = w0 + t1 * (w1 + w2 * t1);
    }

    // ---- stage 1: H[m][n] = sum_x block[m][x] * basisX_n(x) ----
    v8f d1 = {};
#pragma unroll
    for (int i = 0; i < 4; ++i) {
        int x0 = 4 * i + 2 * h;
        v2f a;
        float2 p = *reinterpret_cast<float2*>(&tile[m][cb + x0]);
        a.x = p.x;
        a.y = p.y;
        d1 = __builtin_amdgcn_wmma_f32_16x16x4_f32(false, a, false, bas[i],
                                                   (short)0, d1, false, false);
    }

    // ---- stage 2: M[j][n] = sum_y basisY_j(y) * H[y][n] ----
    // D layout: vgpr r, lanes0-15 => row r; lanes16-31 => row r+8. Need rows
    // {4i+2h, 4i+2h+1} as B rows at this lane: cross-half via shfl_xor(16).
    v8f d2 = {};
#pragma unroll
    for (int i = 0; i < 4; ++i) {
        v2f b2;
        int bi = (i & 1) * 4;
        float e0 = d1[bi + 0], e1 = d1[bi + 1], e2 = d1[bi + 2], e3 = d1[bi + 3];
        float g0 = __shfl_xor(e0, 16, 32);
        float g1 = __shfl_xor(e1, 16, 32);
        float g2 = __shfl_xor(e2, 16, 32);
        float g3 = __shfl_xor(e3, 16, 32);
        if (i < 2) { b2.x = h ? g2 : e0;  b2.y = h ? g3 : e1; }
        else       { b2.x = h ? e2 : g0;  b2.y = h ? e3 : g1; }
        d2 = __builtin_amdgcn_wmma_f32_16x16x4_f32(false, bas[i], false, b2,
                                                   (short)0, d2, false, false);
    }

    // moments live at D[j][n]: vgpr j, lane n
    float S   = __shfl(d2[0], 0, 32);
    float Sx  = __shfl(d2[0], 1, 32);
    float Sxx = __shfl(d2[0], 2, 32);
    float Sy  = __shfl(d2[1], 0, 32);
    float Sxy = __shfl(d2[1], 1, 32);
    float Syy = __shfl(d2[2], 0, 32);

    float T   = S + 1e-8f;
    float inv = 1.0f / T;
    float mux = Sx * inv, muy = Sy * inv;
    float q   = S * inv;
    float cxx = Sxx * inv - mux * mux * (2.0f - q);
    float cyy = Syy * inv - muy * muy * (2.0f - q);
    float cxy = Sxy * inv - mux * muy * (2.0f - q);
    float tr  = cxx + cyy;
    float det = cxx * cyy - cxy * cxy;
    float disc = sqrtf(fmaxf(tr * tr - 4.0f * det, 0.0f));
    float L1 = 0.5f * (tr + disc), L2 = 0.5f * (tr - disc);
    float vx = L1 - cyy, vy = cxy;
    float nrm = fmaxf(sqrtf(vx * vx + vy * vy), 1e-12f);
    vx /= nrm; vy /= nrm;

    if (lane == 0) {
        float* rec = ws + WS_REC + ((size_t)b * NBLK + l) * 8;
        rec[0] = mux; rec[1] = muy; rec[2] = vx; rec[3] = vy;
        rec[4] = L1;  rec[5] = L2;  rec[6] = S * (1.0f / 256.0f);

        // fold block raw moments into per-batch global raw moments
        float x0f = (float)((l & 63) * 16);
        float y0f = (float)((l >> 6) * 16);
        float* acc = ws + WS_ACC + b * 8;
        atomicAdd(acc + 0, S);
        atomicAdd(acc + 1, Sx + x0f * S);
        atomicAdd(acc + 2, Sy + y0f * S);
        atomicAdd(acc + 3, Sxx + 2.0f * x0f * Sx + x0f * x0f * S);
        atomicAdd(acc + 4, Syy + 2.0f * y0f * Sy + y0f * y0f * S);
        atomicAdd(acc + 5, Sxy + x0f * Sy + y0f * Sx + x0f * y0f * S);
    }
}

__global__ __launch_bounds__(256) void k_final(const float* __restrict__ ws,
                                               float* __restrict__ out) {
    const int b   = blockIdx.x;
    const int tid = threadIdx.x;
    __shared__ float sm[3];
    if (tid < 3) sm[tid] = 0.0f;
    __syncthreads();

    // global mean over the whole tensor
    float total = 0.0f;
    for (int bb = 0; bb < 64; ++bb) total += ws[WS_ACC + bb * 8 + 0];
    const float mean = total * (1.0f / 67108864.0f);

    float m00 = 0.0f, m01 = 0.0f, m11 = 0.0f;
    for (int l = tid; l < NBLK; l += 256) {
        const float* rec = ws + WS_REC + ((size_t)b * NBLK + l) * 8;
        float mux = rec[0], muy = rec[1], vx = rec[2], vy = rec[3];
        float L1 = rec[4], L2 = rec[5], bm = rec[6];
        float c0  = (L1 / L2 - 1.0f) * (bm / mean);
        float sg  = 1.0f / (1.0f + expf(-c0));
        float conf = 20.0f * (sg - 0.5f);
        float ox = vx * conf, oy = vy * conf;
        float* o = out + OUT_OUTPUT + ((size_t)b * NBLK + l) * 4;
        o[0] = mux; o[1] = muy; o[2] = ox; o[3] = oy;
        m00 += ox * ox; m01 += ox * oy; m11 += oy * oy;
    }
    atomicAdd(&sm[0], m00);
    atomicAdd(&sm[1], m01);
    atomicAdd(&sm[2], m11);
    __syncthreads();

    if (tid == 0) {
        float M00 = sm[0], M01 = sm[1], M11 = sm[2];
        float T_  = M00 + M11;
        float D   = M00 * M11 - M01 * M01;
        float L1g = 0.5f * (T_ + sqrtf(fmaxf(T_ * T_ - 4.0f * D, 0.0f)));
        float vx = L1g - M11, vy = M01;
        float nrm = fmaxf(sqrtf(vx * vx + vy * vy), 1e-12f);
        vx /= nrm; vy /= nrm;
        out[OUT_U + b * 2 + 0] = vx;
        out[OUT_U + b * 2 + 1] = vy;

        const float* acc = ws + WS_ACC + b * 8;
        float Sg = acc[0], Sxg = acc[1], Syg = acc[2];
        float Tg = Sg + 1e-8f;
        float mux = Sxg / Tg, muy = Syg / Tg;
        out[OUT_MU + b * 2 + 0] = mux;
        out[OUT_MU + b * 2 + 1] = muy;

        const float PI = 3.14159265358979323846f;
        float rot  = fmodf(atan2f(vy, vx + 1e-4f) + 2.0f * PI, PI);
        float diff = 0.5f * PI - rot;
        float c = cosf(diff), s = sinf(diff);
        float mtx = (mux * (1.0f / 1024.0f) - 0.5f) * 2.0f;
        float mty = (muy * (1.0f / 1024.0f) - 0.5f) * (-2.0f);
        float* th = out + OUT_THETA + b * 6;
        th[0] = c;  th[1] = -s; th[2] = mtx;
        th[3] = s;  th[4] = c;  th[5] = mty;
    }
}

extern "C" void kernel_launch(void* const* d_in, const int* in_sizes, int n_in,
                              void* d_out, int out_size, void* d_ws, size_t ws_size,
                              hipStream_t stream) {
    const float* x = (const float*)d_in[0];
    float* out = (float*)d_out;
    float* ws  = (float*)d_ws;

    k_zero_acc<<<2, 256, 0, stream>>>(ws);
    k_blocks<<<64 * 64 * 8, 256, 0, stream>>>(x, out, ws);
    k_final<<<64, 256, 0, stream>>>(ws, out);
}